// VoxelCondEncoder_12094627905854
// MI455X (gfx1250) — compile-verified
//
#include <hip/hip_runtime.h>
#include <hip/hip_bf16.h>

// ---------------------------------------------------------------------------
// CDNA5 (gfx1250, wave32) implementation of VoxelCondEncoder.
//  - fp32 WMMA (V_WMMA_F32_16X16X4_F32) for the PointEmbed GEMM (6.9 GFLOP,
//    the FLOP hotspot) and for the two [*,512]x[512,128] GEMMs (f2 + conv5).
//  - B operands staged transposed in LDS (conflict-free ds_load_b64 fragment
//    reads; zero-padded K removes inner-loop selects).
//  - Direct convs for layers 1-4 (memory-bound; voxel read 134MB ~= 5.8us
//    HBM floor) with global_prefetch priming, and deterministic 2-stage BN
//    batch-stat reductions (no float atomics).
// ---------------------------------------------------------------------------

typedef float v2f __attribute__((ext_vector_type(2)));
typedef float v8f __attribute__((ext_vector_type(8)));

#define PI_F 3.14159265358979323846f

// ---------------------------------------------------------------------------
// PointEmbed value generator: emb = [sin(proj0..23), cos(proj0..23), cx,cy,cz]
// proj[k] = centers[axis] * 2^f * pi, axis=k/8, f=k%8
// ---------------------------------------------------------------------------
__device__ __forceinline__ float emb_val(float cx, float cy, float cz, int k) {
  if (k < 48) {
    int kk = (k < 24) ? k : (k - 24);
    int axis = kk >> 3;
    int f = kk & 7;
    float c = (axis == 0) ? cx : ((axis == 1) ? cy : cz);
    float p = c * (PI_F * (float)(1 << f));
    return (k < 24) ? __sinf(p) : __cosf(p);
  }
  if (k == 48) return cx;
  if (k == 49) return cy;
  if (k == 50) return cz;
  return 0.0f; // zero-pad K=51
}

// ---------------------------------------------------------------------------
// Kernel 1: PointEmbed GEMM via V_WMMA_F32_16X16X4_F32.
//   One wave per 16-point M-tile (B*N/16 = 32768 waves, exact grid).
//   A[16x4] fragment generated in registers (sin/cos on the fly).
//   pe_w staged transposed+zero-padded in LDS: lwT[n][k], k in [0,52).
//   Row stride 52 (= 4*13 mod 64) -> the 16 lanes of each half-wave hit
//   16 distinct LDS bank pairs -> conflict-free ds_load_b64.
//   The f1 (128->1) contraction is fused with half-wave shfl_xor butterflies
//   so the [B,N,128] activation never touches memory.
// ---------------------------------------------------------------------------
__global__ void pe_gemm_wmma_kernel(const float* __restrict__ centers,
                                    const float* __restrict__ pe_w,  // [51,128]
                                    const float* __restrict__ pe_b,  // [128]
                                    const float* __restrict__ f1_w,  // [128]
                                    const float* __restrict__ f1_b,  // [1]
                                    float* __restrict__ s_out) {     // [B*N]
  __shared__ __align__(16) float lwT[128 * 52]; // 26 KB, transposed pe_w

  // Cooperative stage: coalesced reads of pe_w (n contiguous), LDS scatter.
  for (int i = threadIdx.x; i < 52 * 128; i += 256) {
    const int n = i & 127;
    const int k = i >> 7;
    lwT[n * 52 + k] = (k < 51) ? pe_w[k * 128 + n] : 0.0f;
  }
  __syncthreads();

  const int wave = blockIdx.x * (blockDim.x >> 5) + (threadIdx.x >> 5);
  const int lane = threadIdx.x & 31;
  const int m    = lane & 15;   // row within tile / column-in-tile for B,C
  const int half = lane >> 4;   // 0: K=0,1 rows 0-7 ; 1: K=2,3 rows 8-15
  const long long gp = (long long)wave * 16 + m; // global point id

  const float cx = centers[gp * 3 + 0];
  const float cy = centers[gp * 3 + 1];
  const float cz = centers[gp * 3 + 2];

  v8f acc[8] = {};

  for (int kk = 0; kk < 13; ++kk) {
    const int k0 = kk * 4 + half * 2;
    v2f a;
    a.x = emb_val(cx, cy, cz, k0);
    a.y = emb_val(cx, cy, cz, k0 + 1);
#pragma unroll
    for (int t = 0; t < 8; ++t) {
      const int n = t * 16 + m;
      const v2f b = *(const v2f*)&lwT[n * 52 + k0]; // ds_load_b64
      acc[t] = __builtin_amdgcn_wmma_f32_16x16x4_f32(
          false, a, false, b, (short)0, acc[t], false, false);
    }
  }

  // Fused f1: s[row] = f1_b + sum_n (xc[row,n] + pe_b[n]) * f1_w[n]
  const float fb = f1_b[0];
  float partial[8];
#pragma unroll
  for (int r = 0; r < 8; ++r) {
    float p = 0.0f;
#pragma unroll
    for (int t = 0; t < 8; ++t) {
      const int n = t * 16 + m;
      p += (acc[t][r] + pe_b[n]) * f1_w[n];
    }
    // reduce over the 16 lanes of this half-wave (xor of low 4 bits only)
#pragma unroll
    for (int off = 8; off >= 1; off >>= 1) p += __shfl_xor(p, off, 32);
    partial[r] = p;
  }
  // writer lane m (< 8) emits row  8*half + m  of this tile
  const int rsel = lane & 7;
  float v = partial[0];
#pragma unroll
  for (int r = 1; r < 8; ++r) v = (rsel == r) ? partial[r] : v;
  if (m < 8) s_out[(long long)wave * 16 + 8 * half + rsel] = v + fb;
}

// ---------------------------------------------------------------------------
// Kernel 2: WMMA fp32 GEMM, C[M,128] = A[M,512] @ Bw[512,128] + bias.
//   Used for both the f2 projection and conv layer 5 (as packed GEMM).
//   One wave per 16-row M-tile; B staged transposed in LDS in four K-slabs
//   of 128 (row stride 130 -> 2 mod 64 -> conflict-free ds_load_b64).
// ---------------------------------------------------------------------------
__global__ void gemm512_wmma_kernel(const float* __restrict__ A,
                                    const float* __restrict__ Bw,
                                    const float* __restrict__ bias,
                                    float* __restrict__ C) {
  __shared__ __align__(16) float lb[128 * 130]; // 65 KB transposed K-slab

  const int wave = blockIdx.x * (blockDim.x >> 5) + (threadIdx.x >> 5);
  const int lane = threadIdx.x & 31;
  const int m    = lane & 15;
  const int half = lane >> 4;
  const float* arow = A + ((long long)(wave * 16 + m)) * 512;

  v8f acc[8] = {};

  for (int slab = 0; slab < 4; ++slab) {
    __syncthreads(); // previous slab fully consumed
    // Stage Bw[slab*128 .. +128)[0..128) transposed: lb[n*130 + k]
    for (int i = threadIdx.x; i < 128 * 128; i += 256) {
      const int n = i & 127;               // contiguous global reads
      const int k = i >> 7;
      lb[n * 130 + k] = Bw[(slab * 128 + k) * 128 + n];
    }
    __syncthreads();

    for (int kk = 0; kk < 128; kk += 4) {
      const int k0 = kk + half * 2;
      const v2f a = *(const v2f*)(arow + slab * 128 + k0); // 8B-aligned
#pragma unroll
      for (int t = 0; t < 8; ++t) {
        const int n = t * 16 + m;
        const v2f b = *(const v2f*)&lb[n * 130 + k0];      // ds_load_b64
        acc[t] = __builtin_amdgcn_wmma_f32_16x16x4_f32(
            false, a, false, b, (short)0, acc[t], false, false);
      }
    }
  }

#pragma unroll
  for (int r = 0; r < 8; ++r) {
    const int orow = wave * 16 + 8 * half + r;
#pragma unroll
    for (int t = 0; t < 8; ++t) {
      const int n = t * 16 + m;
      C[(long long)orow * 128 + n] = acc[t][r] + bias[n];
    }
  }
}

// ---------------------------------------------------------------------------
// Direct conv3d, kernel 3, stride 2, pad 1 (layers 1..4). Thread per output.
// ---------------------------------------------------------------------------
__global__ void conv3d_s2_kernel(const float* __restrict__ x,
                                 const float* __restrict__ W,
                                 const float* __restrict__ bias,
                                 float* __restrict__ y,
                                 int Bn, int Cin, int Cout, int Din, int Dout) {
  const long long total = (long long)Bn * Cout * Dout * Dout * Dout;
  const long long idx = (long long)blockIdx.x * blockDim.x + threadIdx.x;
  if (idx >= total) return;
  long long t = idx;
  const int ox = (int)(t % Dout); t /= Dout;
  const int oy = (int)(t % Dout); t /= Dout;
  const int oz = (int)(t % Dout); t /= Dout;
  const int oc = (int)(t % Cout);
  const int b  = (int)(t / Cout);

  float acc = bias[oc];
  const int D2 = Din * Din;
  const float* wbase = W + (long long)oc * Cin * 27;
  for (int ic = 0; ic < Cin; ++ic) {
    const float* xin = x + ((long long)(b * Cin + ic)) * Din * D2;
    // Prime the load pipe for the next input channel (global_prefetch_b8).
    if (ic + 1 < Cin)
      __builtin_prefetch(xin + (long long)Din * D2, 0, 1);
    const float* wk  = wbase + ic * 27;
#pragma unroll
    for (int kz = 0; kz < 3; ++kz) {
      const int iz = 2 * oz - 1 + kz;
      if ((unsigned)iz >= (unsigned)Din) continue;
#pragma unroll
      for (int ky = 0; ky < 3; ++ky) {
        const int iy = 2 * oy - 1 + ky;
        if ((unsigned)iy >= (unsigned)Din) continue;
#pragma unroll
        for (int kx = 0; kx < 3; ++kx) {
          const int ix = 2 * ox - 1 + kx;
          if ((unsigned)ix >= (unsigned)Din) continue;
          acc += xin[(long long)iz * D2 + iy * Din + ix] * wk[kz * 9 + ky * 3 + kx];
        }
      }
    }
  }
  y[idx] = acc;
}

// ---------------------------------------------------------------------------
// BN batch-stat reduction, stage 1: per (channel, split) partial sum / sumsq.
// Deterministic tree reduction (no float atomics). blockDim = 256.
// ---------------------------------------------------------------------------
__global__ void bn_partial_kernel(const float* __restrict__ y,
                                  float* __restrict__ part,
                                  int Bn, int C, int D3, int NS) {
  const int c    = blockIdx.x / NS;
  const int sidx = blockIdx.x % NS;
  const long long E = (long long)Bn * D3;
  float s = 0.0f, s2 = 0.0f;
  for (long long e = (long long)sidx * 256 + threadIdx.x; e < E;
       e += (long long)NS * 256) {
    const int b   = (int)(e / D3);
    const int off = (int)(e % D3);
    const float v = y[((long long)b * C + c) * D3 + off];
    s += v;
    s2 += v * v;
  }
  __shared__ float sh[512];
  sh[threadIdx.x]       = s;
  sh[threadIdx.x + 256] = s2;
  __syncthreads();
  for (int w = 128; w > 0; w >>= 1) {
    if ((int)threadIdx.x < w) {
      sh[threadIdx.x]       += sh[threadIdx.x + w];
      sh[threadIdx.x + 256] += sh[threadIdx.x + 256 + w];
    }
    __syncthreads();
  }
  if (threadIdx.x == 0) {
    part[(c * NS + sidx) * 2 + 0] = sh[0];
    part[(c * NS + sidx) * 2 + 1] = sh[256];
  }
}

// Stage 2: fold partials -> per-channel affine (a, b): y' = y*a + b
__global__ void bn_finalize_kernel(const float* __restrict__ part,
                                   const float* __restrict__ g,
                                   const float* __restrict__ beta,
                                   float* __restrict__ stats,
                                   int C, int NS, float invE) {
  const int c = blockIdx.x * blockDim.x + threadIdx.x;
  if (c >= C) return;
  float s = 0.0f, s2 = 0.0f;
  for (int i = 0; i < NS; ++i) {
    s  += part[(c * NS + i) * 2 + 0];
    s2 += part[(c * NS + i) * 2 + 1];
  }
  const float mean = s * invE;
  const float var  = s2 * invE - mean * mean;
  const float a    = rsqrtf(var + 1e-5f) * g[c];
  stats[c * 2 + 0] = a;
  stats[c * 2 + 1] = beta[c] - mean * a;
}

// In-place BN scale/shift + ReLU
__global__ void bn_relu_kernel(float* __restrict__ y,
                               const float* __restrict__ stats,
                               long long total, int C, int D3) {
  const long long idx = (long long)blockIdx.x * blockDim.x + threadIdx.x;
  if (idx >= total) return;
  const int c = (int)((idx / D3) % C);
  const float v = y[idx] * stats[c * 2 + 0] + stats[c * 2 + 1];
  y[idx] = fmaxf(v, 0.0f);
}

// Pack W5[128,64,3,3,3] -> Wg[512,128]: only taps k in {1,2} survive for the
// 2^3 -> 1^3 stride-2 pad-1 conv, so layer 5 is exactly a K=512 GEMM.
__global__ void prep_wg_kernel(const float* __restrict__ W5,
                               float* __restrict__ wg) {
  const int idx = blockIdx.x * blockDim.x + threadIdx.x;
  if (idx >= 512 * 128) return;
  const int oc = idx & 127;
  const int k  = idx >> 7;
  const int ic = k >> 3;
  const int dz = (k >> 2) & 1;
  const int dy = (k >> 1) & 1;
  const int dx = k & 1;
  wg[idx] = W5[((((long long)oc * 64 + ic) * 3 + (1 + dz)) * 3 + (1 + dy)) * 3 + (1 + dx)];
}

// Final: out = relu(bn(y5)) + xc   over [1024,128]
__global__ void final_add_kernel(const float* __restrict__ y5,
                                 const float* __restrict__ stats,
                                 const float* __restrict__ xc,
                                 float* __restrict__ out) {
  const int idx = blockIdx.x * blockDim.x + threadIdx.x;
  if (idx >= 1024 * 128) return;
  const int c = idx & 127;
  const float v = y5[idx] * stats[c * 2 + 0] + stats[c * 2 + 1];
  out[idx] = fmaxf(v, 0.0f) + xc[idx];
}

// ---------------------------------------------------------------------------
// Host launch
// ---------------------------------------------------------------------------
extern "C" void kernel_launch(void* const* d_in, const int* in_sizes, int n_in,
                              void* d_out, int out_size, void* d_ws, size_t ws_size,
                              hipStream_t stream) {
  const float* voxels  = (const float*)d_in[0];
  const float* centers = (const float*)d_in[1];
  const float* Wc[5], *bc[5], *gc[5], *betac[5];
  for (int i = 0; i < 5; ++i) {
    Wc[i]    = (const float*)d_in[2 + 4 * i];
    bc[i]    = (const float*)d_in[3 + 4 * i];
    gc[i]    = (const float*)d_in[4 + 4 * i];
    betac[i] = (const float*)d_in[5 + 4 * i];
  }
  const float* pe_w = (const float*)d_in[22];
  const float* pe_b = (const float*)d_in[23];
  const float* f1_w = (const float*)d_in[24];
  const float* f1_b = (const float*)d_in[25];
  const float* f2_w = (const float*)d_in[26];
  const float* f2_b = (const float*)d_in[27];
  float* out = (float*)d_out;

  // Workspace carve-up (ping-pong conv buffers): ~104 MB total
  char* ws = (char*)d_ws;
  float* bufA  = (float*)(ws);                          // 64 MiB (L1 out / L3 out)
  float* bufB  = (float*)(ws + (size_t)67108864);       // 32 MiB (L2 out / L4 out)
  float* sbuf  = (float*)(ws + (size_t)100663296);      // 2 MiB  s[B*N]
  float* xcbuf = (float*)(ws + (size_t)102760448);      // 512 KiB xc[B,128]
  float* y5buf = (float*)(ws + (size_t)103284736);      // 512 KiB y5[B,128]
  float* wgbuf = (float*)(ws + (size_t)103809024);      // 256 KiB Wg[512,128]
  float* part  = (float*)(ws + (size_t)104071168);      // 8 KiB partial stats
  float* stats = (float*)(ws + (size_t)104079360);      // 1 KiB per-channel affine

  // ---- PointEmbed branch -------------------------------------------------
  // 32768 M-tiles, 8 waves/block -> exact grid (EXEC all-ones for WMMA)
  pe_gemm_wmma_kernel<<<4096, 256, 0, stream>>>(centers, pe_w, pe_b, f1_w, f1_b, sbuf);
  // xc[1024,128] = s[1024,512] @ f2_w + f2_b : 64 waves, 8 blocks exact
  gemm512_wmma_kernel<<<8, 256, 0, stream>>>(sbuf, f2_w, f2_b, xcbuf);

  // ---- Conv stack layers 1..4 -------------------------------------------
  const int   Cin_[4]  = {1, 4, 16, 32};
  const int   Cout_[4] = {4, 16, 32, 64};
  const int   Din_[4]  = {32, 16, 8, 4};
  const float* src[4]  = {voxels, bufA, bufB, bufA};
  float*       dst[4]  = {bufA, bufB, bufA, bufB};
  for (int i = 0; i < 4; ++i) {
    const int Dout = Din_[i] >> 1;
    const int D3   = Dout * Dout * Dout;
    const long long total = (long long)1024 * Cout_[i] * D3;
    const int NS = 1024 / Cout_[i]; // keeps C*NS == 1024 partial entries
    conv3d_s2_kernel<<<(unsigned)((total + 255) / 256), 256, 0, stream>>>(
        src[i], Wc[i], bc[i], dst[i], 1024, Cin_[i], Cout_[i], Din_[i], Dout);
    bn_partial_kernel<<<Cout_[i] * NS, 256, 0, stream>>>(dst[i], part, 1024, Cout_[i], D3, NS);
    bn_finalize_kernel<<<1, 128, 0, stream>>>(part, gc[i], betac[i], stats,
                                              Cout_[i], NS, 1.0f / (float)(1024LL * D3));
    bn_relu_kernel<<<(unsigned)((total + 255) / 256), 256, 0, stream>>>(
        dst[i], stats, total, Cout_[i], D3);
  }

  // ---- Conv layer 5 as WMMA GEMM ----------------------------------------
  prep_wg_kernel<<<256, 256, 0, stream>>>(Wc[4], wgbuf);
  // x4flat [1024,512] lives in bufB ([B,64,2,2,2] contiguous == ic*8+dz*4+dy*2+dx)
  gemm512_wmma_kernel<<<8, 256, 0, stream>>>(bufB, wgbuf, bc[4], y5buf);
  bn_partial_kernel<<<128 * 8, 256, 0, stream>>>(y5buf, part, 1024, 128, 1, 8);
  bn_finalize_kernel<<<1, 128, 0, stream>>>(part, gc[4], betac[4], stats, 128, 8,
                                            1.0f / 1024.0f);

  // ---- Fuse: out = relu(bn(y5)) + xc ------------------------------------
  final_add_kernel<<<512, 256, 0, stream>>>(y5buf, stats, xcbuf, out);
}